// MuellerMatrixModel_86517821210793
// MI455X (gfx1250) — compile-verified
//
#include <hip/hip_runtime.h>
#include <math.h>

// ---------------------------------------------------------------------------
// Problem constants (fixed by the reference): x is (B=2, F=48, H=512, W=512)
// output y is (B, 34, H, W):  [0:16]=intensity copy, [16:32]=Mueller M,
// [32]=azimuth, [33]=circular std over 4x4 SAME window.
// ---------------------------------------------------------------------------
#define BATCH 2
#define HGT   512
#define WID   512
#define HW    (HGT * WID)          // 262144
#define OUTC  34
#define PI_F  3.14159265358979323846f

// ---- CDNA5 async global->LDS path (guarded; falls back to plain loads) ----
#define AS_GLOBAL __attribute__((address_space(1)))
#define AS_LOCAL  __attribute__((address_space(3)))

#if defined(__has_builtin)
#  if __has_builtin(__builtin_amdgcn_global_load_async_to_lds_b32)
#    define HAVE_ASYNC_LDS 1
#  endif
#endif

__device__ __forceinline__ void wait_async_lds() {
#if defined(__has_builtin)
#  if __has_builtin(__builtin_amdgcn_s_wait_asynccnt)
  __builtin_amdgcn_s_wait_asynccnt(0);
  return;
#  endif
#endif
  asm volatile("s_wait_asynccnt 0" ::: "memory");
}

// ---------------------------------------------------------------------------
// Kernel 2 (defined first so it leads the disassembly): 4x4 SAME-window
// circular std of 2*azimuth. 32x8 output tile per block with a (-1,+2) halo
// -> 11x35 LDS tile. The azimuth halo tile is staged into LDS via CDNA5
// async global->LDS loads (ASYNCcnt), then converted to cos/sin planes.
// Window: rows/cols [i-1, i+2] (JAX 'SAME' pads lo=1, hi=2 for window 4);
// out-of-image taps contribute 0 and are excluded from the count.
// ---------------------------------------------------------------------------
#define TR 11            // tile rows  (8 + 3 halo)
#define TC 35            // tile cols  (32 + 3 halo)
#define TCP 36           // padded LDS row stride (avoid bank conflicts)

__global__ __launch_bounds__(256)
void std_kernel(const float* __restrict__ y_in, float* __restrict__ y_out) {
  __shared__ float sa[TR * TCP];   // raw azimuth halo tile
  __shared__ float sc[TR * TCP];   // cos(2*azi), 0 outside image
  __shared__ float ss[TR * TCP];   // sin(2*azi), 0 outside image

  const int b  = blockIdx.z;
  const int h0 = blockIdx.y * 8;
  const int w0 = blockIdx.x * 32;
  const int tid = threadIdx.y * 32 + threadIdx.x;

  const float* azi = y_in + ((size_t)b * OUTC + 32) * HW;

  // Stage haloed azimuth tile into LDS (async DMA path on CDNA5).
  for (int t = tid; t < TR * TC; t += 256) {
    int r = t / TC, c = t % TC;
    int gh = h0 - 1 + r;
    int gw = w0 - 1 + c;
    if ((unsigned)gh < (unsigned)HGT && (unsigned)gw < (unsigned)WID) {
#if defined(HAVE_ASYNC_LDS)
      __builtin_amdgcn_global_load_async_to_lds_b32(
          (AS_GLOBAL int*)(azi + (size_t)gh * WID + gw),
          (AS_LOCAL int*)&sa[r * TCP + c], 0, 0);
#else
      sa[r * TCP + c] = azi[(size_t)gh * WID + gw];
#endif
    }
  }
#if defined(HAVE_ASYNC_LDS)
  wait_async_lds();
#endif
  __syncthreads();

  // Convert to cos/sin planes (zero outside the image).
  for (int t = tid; t < TR * TC; t += 256) {
    int r = t / TC, c = t % TC;
    int gh = h0 - 1 + r;
    int gw = w0 - 1 + c;
    float cv = 0.0f, sv = 0.0f;
    if ((unsigned)gh < (unsigned)HGT && (unsigned)gw < (unsigned)WID) {
      float a2 = 2.0f * sa[r * TCP + c];
      sincosf(a2, &sv, &cv);
    }
    sc[r * TCP + c] = cv;
    ss[r * TCP + c] = sv;
  }
  __syncthreads();

  const int h = h0 + threadIdx.y;
  const int w = w0 + threadIdx.x;

  float cs = 0.0f, sn = 0.0f;
#pragma unroll
  for (int dr = 0; dr < 4; ++dr) {
    int base = (threadIdx.y + dr) * TCP + threadIdx.x;
#pragma unroll
    for (int dc = 0; dc < 4; ++dc) {
      cs += sc[base + dc];
      sn += ss[base + dc];
    }
  }

  // Edge-count normalization (separable counts).
  int lo_h = max(h - 1, 0), hi_h = min(h + 2, HGT - 1);
  int lo_w = max(w - 1, 0), hi_w = min(w + 2, WID - 1);
  float invcnt = 1.0f / (float)((hi_h - lo_h + 1) * (hi_w - lo_w + 1));

  float cm = cs * invcnt;
  float sm = sn * invcnt;
  float R = sqrtf(cm * cm + sm * sm);
  R = fminf(fmaxf(R, 1e-7f), 1.0f - 1e-7f);
  float stdv = 0.5f * sqrtf(-2.0f * logf(R));

  y_out[((size_t)b * OUTC + 33) * HW + (size_t)h * WID + w] = stdv;
}

// ---------------------------------------------------------------------------
// 4x4 inverse via adjugate/cofactors. The classic formulation is
// layout-agnostic: feeding a row-major matrix returns its row-major inverse.
// ---------------------------------------------------------------------------
__device__ __forceinline__ void inv4(const float m[16], float o[16]) {
  o[0]  =  m[5]*m[10]*m[15] - m[5]*m[11]*m[14] - m[9]*m[6]*m[15]
         + m[9]*m[7]*m[14] + m[13]*m[6]*m[11] - m[13]*m[7]*m[10];
  o[4]  = -m[4]*m[10]*m[15] + m[4]*m[11]*m[14] + m[8]*m[6]*m[15]
         - m[8]*m[7]*m[14] - m[12]*m[6]*m[11] + m[12]*m[7]*m[10];
  o[8]  =  m[4]*m[9]*m[15]  - m[4]*m[11]*m[13] - m[8]*m[5]*m[15]
         + m[8]*m[7]*m[13]  + m[12]*m[5]*m[11] - m[12]*m[7]*m[9];
  o[12] = -m[4]*m[9]*m[14]  + m[4]*m[10]*m[13] + m[8]*m[5]*m[14]
         - m[8]*m[6]*m[13]  - m[12]*m[5]*m[10] + m[12]*m[6]*m[9];
  o[1]  = -m[1]*m[10]*m[15] + m[1]*m[11]*m[14] + m[9]*m[2]*m[15]
         - m[9]*m[3]*m[14]  - m[13]*m[2]*m[11] + m[13]*m[3]*m[10];
  o[5]  =  m[0]*m[10]*m[15] - m[0]*m[11]*m[14] - m[8]*m[2]*m[15]
         + m[8]*m[3]*m[14]  + m[12]*m[2]*m[11] - m[12]*m[3]*m[10];
  o[9]  = -m[0]*m[9]*m[15]  + m[0]*m[11]*m[13] + m[8]*m[1]*m[15]
         - m[8]*m[3]*m[13]  - m[12]*m[1]*m[11] + m[12]*m[3]*m[9];
  o[13] =  m[0]*m[9]*m[14]  - m[0]*m[10]*m[13] - m[8]*m[1]*m[14]
         + m[8]*m[2]*m[13]  + m[12]*m[1]*m[10] - m[12]*m[2]*m[9];
  o[2]  =  m[1]*m[6]*m[15]  - m[1]*m[7]*m[14]  - m[5]*m[2]*m[15]
         + m[5]*m[3]*m[14]  + m[13]*m[2]*m[7]  - m[13]*m[3]*m[6];
  o[6]  = -m[0]*m[6]*m[15]  + m[0]*m[7]*m[14]  + m[4]*m[2]*m[15]
         - m[4]*m[3]*m[14]  - m[12]*m[2]*m[7]  + m[12]*m[3]*m[6];
  o[10] =  m[0]*m[5]*m[15]  - m[0]*m[7]*m[13]  - m[4]*m[1]*m[15]
         + m[4]*m[3]*m[13]  + m[12]*m[1]*m[7]  - m[12]*m[3]*m[5];
  o[14] = -m[0]*m[5]*m[14]  + m[0]*m[6]*m[13]  + m[4]*m[1]*m[14]
         - m[4]*m[2]*m[13]  - m[12]*m[1]*m[6]  + m[12]*m[2]*m[5];
  o[3]  = -m[1]*m[6]*m[11]  + m[1]*m[7]*m[10]  + m[5]*m[2]*m[11]
         - m[5]*m[3]*m[10]  - m[9]*m[2]*m[7]   + m[9]*m[3]*m[6];
  o[7]  =  m[0]*m[6]*m[11]  - m[0]*m[7]*m[10]  - m[4]*m[2]*m[11]
         + m[4]*m[3]*m[10]  + m[8]*m[2]*m[7]   - m[8]*m[3]*m[6];
  o[11] = -m[0]*m[5]*m[11]  + m[0]*m[7]*m[9]   + m[4]*m[1]*m[11]
         - m[4]*m[3]*m[9]   - m[8]*m[1]*m[7]   + m[8]*m[3]*m[5];
  o[15] =  m[0]*m[5]*m[10]  - m[0]*m[6]*m[9]   - m[4]*m[1]*m[10]
         + m[4]*m[2]*m[9]   + m[8]*m[1]*m[6]   - m[8]*m[2]*m[5];
  float det = m[0]*o[0] + m[1]*o[4] + m[2]*o[8] + m[3]*o[12];
  float rd = 1.0f / det;
#pragma unroll
  for (int i = 0; i < 16; ++i) o[i] *= rd;
}

__device__ __forceinline__ void mm4(const float a[16], const float b[16],
                                    float c[16]) {
#pragma unroll
  for (int i = 0; i < 4; ++i)
#pragma unroll
    for (int j = 0; j < 4; ++j) {
      float acc = a[i * 4 + 0] * b[0 * 4 + j];
      acc = fmaf(a[i * 4 + 1], b[1 * 4 + j], acc);
      acc = fmaf(a[i * 4 + 2], b[2 * 4 + j], acc);
      acc = fmaf(a[i * 4 + 3], b[3 * 4 + j], acc);
      c[i * 4 + j] = acc;
    }
}

// ---------------------------------------------------------------------------
// Kernel 1: per-pixel Mueller decomposition + azimuth.
// One thread per pixel; all channel loads/stores are plane-contiguous
// (coalesced fp32 streams) -> this kernel runs at memory bandwidth.
// ---------------------------------------------------------------------------
__global__ __launch_bounds__(256)
void mueller_kernel(const float* __restrict__ x, float* __restrict__ y) {
  int idx = blockIdx.x * blockDim.x + threadIdx.x;
  if (idx >= BATCH * HW) return;
  int b = idx >> 18;          // HW == 2^18
  int p = idx & (HW - 1);

  const float* xb = x + (size_t)b * 48 * HW + p;
  float I[16], A[16], Wm[16];
#pragma unroll
  for (int c = 0; c < 16; ++c) I[c]  = xb[(size_t)c * HW];
#pragma unroll
  for (int c = 0; c < 16; ++c) A[c]  = xb[(size_t)(16 + c) * HW];
#pragma unroll
  for (int c = 0; c < 16; ++c) Wm[c] = xb[(size_t)(32 + c) * HW];

  float iA[16], iW[16], T[16], M[16];
  inv4(A, iA);
  inv4(Wm, iW);
  mm4(iA, I, T);
  mm4(T, iW, M);

  float* yb = y + (size_t)b * OUTC * HW + p;
#pragma unroll
  for (int c = 0; c < 16; ++c) yb[(size_t)c * HW]        = I[c];
#pragma unroll
  for (int c = 0; c < 16; ++c) yb[(size_t)(16 + c) * HW] = M[c];

  // azimuth in [0, pi):  0.5*atan2(M[3][1], -M[3][2]) mod pi
  float azi = 0.5f * atan2f(M[13], -M[14]);
  if (azi < 0.0f) azi += PI_F;
  yb[(size_t)32 * HW] = azi;
}

// ---------------------------------------------------------------------------
extern "C" void kernel_launch(void* const* d_in, const int* in_sizes, int n_in,
                              void* d_out, int out_size, void* d_ws,
                              size_t ws_size, hipStream_t stream) {
  (void)in_sizes; (void)n_in; (void)out_size; (void)d_ws; (void)ws_size;
  const float* x = (const float*)d_in[0];
  float* y = (float*)d_out;

  // Kernel 1: one thread per pixel (B*H*W = 524288 -> 2048 blocks of 256).
  mueller_kernel<<<(BATCH * HW + 255) / 256, 256, 0, stream>>>(x, y);

  // Kernel 2: 32x8 tiles over (W, H, B).
  dim3 grid(WID / 32, HGT / 8, BATCH);
  dim3 block(32, 8, 1);
  std_kernel<<<grid, block, 0, stream>>>(y, y);
}